// LiveNet_60601988546682
// MI455X (gfx1250) — compile-verified
//
#include <hip/hip_runtime.h>

// WMMA vector types (fp32 path: A/B = 2 VGPRs, C/D = 8 VGPRs per lane, wave32)
typedef float v2f __attribute__((ext_vector_type(2)));
typedef float v8f __attribute__((ext_vector_type(8)));

#define WAVES_PER_BLOCK 8
#define TILE_M 16
#define H_STRIDE 36   // 32 cols + pad: conflict-free LDS, row byte-offset multiple of 8

__global__ __launch_bounds__(256) void mlp_wmma_f32(
    const float* __restrict__ x,   // [B,64]
    const float* __restrict__ W1,  // [64,32]
    const float* __restrict__ b1,  // [32]
    const float* __restrict__ W2,  // [32,16]
    const float* __restrict__ b2,  // [16]
    float* __restrict__ y,         // [B,16]
    int numTiles)                  // B/16
{
    __shared__ float lds_h[WAVES_PER_BLOCK][TILE_M][H_STRIDE];

    const int lane        = threadIdx.x & 31;
    const int waveInBlock = threadIdx.x >> 5;
    const int waveGlobal  = blockIdx.x * WAVES_PER_BLOCK + waveInBlock;
    const int numWaves    = gridDim.x * WAVES_PER_BLOCK;

    const int r    = lane & 15;       // M (A-frag row) / N (B-frag col) within tile
    const int hi   = lane >> 4;       // lane half: 0 = lanes 0-15, 1 = lanes 16-31
    const int koff = hi * 2;          // K offset contributed by the upper lane half

    // ---- Preload W1 as WMMA B-fragments (4x16 K x N chunks).
    // Layout mirrors ISA A16x4 table: VGPR0 = rows K=k0 (low lanes) / k0+2 (high),
    //                                 VGPR1 = rows K=k0+1 / k0+3, N striped over r.
    v2f w1f[16][2];
#pragma unroll
    for (int kk = 0; kk < 16; ++kk) {
#pragma unroll
        for (int nt = 0; nt < 2; ++nt) {
            const int n  = nt * 16 + r;
            const int k0 = kk * 4 + koff;
            w1f[kk][nt].x = W1[(k0 + 0) * 32 + n];
            w1f[kk][nt].y = W1[(k0 + 1) * 32 + n];
        }
    }
    v2f w2f[8];
#pragma unroll
    for (int kk = 0; kk < 8; ++kk) {
        const int k0 = kk * 4 + koff;
        w2f[kk].x = W2[(k0 + 0) * 16 + r];
        w2f[kk].y = W2[(k0 + 1) * 16 + r];
    }
    const float bias1_0 = b1[r];
    const float bias1_1 = b1[16 + r];
    const float bias2   = b2[r];

    float (*hbuf)[H_STRIDE] = lds_h[waveInBlock];

    for (int tile = waveGlobal; tile < numTiles; tile += numWaves) {
        const float* xrow = x + (size_t)tile * TILE_M * 64;

        // ---- GEMM1: h[16,32] = x[16,64] @ W1[64,32]
        v8f acc0 = {};  // N = 0..15
        v8f acc1 = {};  // N = 16..31
#pragma unroll
        for (int kk = 0; kk < 16; ++kk) {
            // A fragment 16x4: per-lane contiguous 8B load (b64)
            const float* p = xrow + r * 64 + kk * 4 + koff;
            v2f a;
            a.x = p[0];
            a.y = p[1];
            acc0 = __builtin_amdgcn_wmma_f32_16x16x4_f32(
                false, a, false, w1f[kk][0], (short)0, acc0, false, false);
            acc1 = __builtin_amdgcn_wmma_f32_16x16x4_f32(
                false, a, false, w1f[kk][1], (short)0, acc1, false, false);
        }

        // ---- bias + ReLU, stage h to LDS (transpose C-layout -> A-layout)
        // C layout: VGPR j holds h[j][n] (lanes 0-15) and h[j+8][n] (lanes 16-31)
#pragma unroll
        for (int j = 0; j < 8; ++j) {
            float h0 = acc0[j] + bias1_0; h0 = h0 > 0.0f ? h0 : 0.0f;
            float h1 = acc1[j] + bias1_1; h1 = h1 > 0.0f ? h1 : 0.0f;
            const int row = j + hi * 8;
            hbuf[row][r]      = h0;
            hbuf[row][16 + r] = h1;
        }
        // Same-wave DS RAW: DS pipe is in-order, but force issue-order + wait.
        asm volatile("s_wait_dscnt 0x0" ::: "memory");

        // ---- GEMM2: y[16,16] = h[16,32] @ W2[32,16]
        v8f accy = {};
#pragma unroll
        for (int kk = 0; kk < 8; ++kk) {
            v2f a;
            a.x = hbuf[r][kk * 4 + koff + 0];
            a.y = hbuf[r][kk * 4 + koff + 1];
            accy = __builtin_amdgcn_wmma_f32_16x16x4_f32(
                false, a, false, w2f[kk], (short)0, accy, false, false);
        }

        // ---- +b2 and store y tile
        float* yrow = y + (size_t)tile * TILE_M * 16;
#pragma unroll
        for (int j = 0; j < 8; ++j) {
            yrow[(j + hi * 8) * 16 + r] = accy[j] + bias2;
        }
        asm volatile("s_wait_dscnt 0x0" ::: "memory"); // keep next tile's LDS stores behind loads
    }
}

extern "C" void kernel_launch(void* const* d_in, const int* in_sizes, int n_in,
                              void* d_out, int out_size, void* d_ws, size_t ws_size,
                              hipStream_t stream) {
    const float* x  = (const float*)d_in[0];
    const float* W1 = (const float*)d_in[1];
    const float* b1 = (const float*)d_in[2];
    const float* W2 = (const float*)d_in[3];
    const float* b2 = (const float*)d_in[4];
    float* y = (float*)d_out;

    const int numTiles = in_sizes[0] / (64 * TILE_M);  // B / 16

    int blocks = 1024;
    const int maxBlocks = (numTiles + WAVES_PER_BLOCK - 1) / WAVES_PER_BLOCK;
    if (blocks > maxBlocks) blocks = maxBlocks;
    if (blocks < 1) blocks = 1;

    mlp_wmma_f32<<<blocks, 256, 0, stream>>>(x, W1, b1, W2, b2, y, numTiles);
}